// RegLoss_46858093200031
// MI455X (gfx1250) — compile-verified
//
#include <hip/hip_runtime.h>

// Problem constants (fixed by the reference): B=32, A=262144, K=131072
#define KK   131072
#define NROWS (32 * 262144)      // B*A = 8,388,608 rows
#define LOG2A 18                 // A = 262144 = 2^18 ; also 2K = 2^18

typedef __attribute__((ext_vector_type(2))) float v2f;
typedef __attribute__((ext_vector_type(4))) float v4f;
typedef __attribute__((ext_vector_type(8))) float v8f;

// Full-precision f32 wave(32) reduction through the WMMA engine.
// A-matrix 16x4 layout (ISA 7.12.2): lanes 0-15 VGPR0=K0/VGPR1=K1 (M=lane),
// lanes 16-31 VGPR0=K2/VGPR1=K3 (M=lane-16).  With a=(x,0) and B = all-ones
// (layout-invariant), D[M,N] = x_M + x_{M+16} for every N.
// C/D layout: lane 0 holds D[0..7,0] in c[0..7], lane 16 holds D[8..15,0].
// => sum = (sum of c in lane 0) + (sum of c in lane 16).  Deterministic, exact order.
__device__ __forceinline__ float wave_sum_f32_wmma(float x) {
  v2f a; a[0] = x;    a[1] = 0.0f;
  v2f b; b[0] = 1.0f; b[1] = 1.0f;
  v8f c = {};
  c = __builtin_amdgcn_wmma_f32_16x16x4_f32(
      /*neg_a=*/false, a, /*neg_b=*/false, b,
      /*c_mod=*/(short)0, c, /*reuse_a=*/false, /*reuse_b=*/false);
  float s = ((c[0] + c[1]) + (c[2] + c[3])) + ((c[4] + c[5]) + (c[6] + c[7]));
  return __shfl(s, 0, 32) + __shfl(s, 16, 32);
}

__device__ __forceinline__ float smooth_l1(float d) {
  float ad = fabsf(d);
  return (ad < 1.0f) ? 0.5f * d * d : ad - 0.5f;
}

__global__ void __launch_bounds__(256) poly_loss_partial(
    const float* __restrict__ input,     // (B, 2K)
    const float* __restrict__ target,    // (B, A, 4)
    float* __restrict__ ws)              // (gridDim.x, 2) partials
{
  const int tid    = blockIdx.x * blockDim.x + threadIdx.x;
  const int stride = gridDim.x * blockDim.x;

  const v4f* __restrict__ t4 = reinterpret_cast<const v4f*>(target);

  float loss = 0.0f;
  float cnt  = 0.0f;

  for (int i = tid; i < NROWS; i += stride) {
    // Streaming read-once data: non-temporal, fully coalesced 16B/lane.
    v4f t = __builtin_nontemporal_load(&t4[i]);
    if (t[3] == 1.0f) {                       // mask: state == 1.0 exactly
      int b = i >> LOG2A;                     // i / A
      int k = (int)t[2];
      k = (k < 0) ? 0 : ((k > KK - 1) ? KK - 1 : k);
      const float* __restrict__ base = input + ((long)b << LOG2A);  // b * 2K
      float gx = base[k];                     // regression[b, 0, k]
      float gy = base[KK + k];                // regression[b, 1, k]
      loss += smooth_l1(gx - t[0]);
      loss += smooth_l1(gy - t[1]);
      cnt  += 1.0f;
    }
  }

  // All 256 threads reach here (EXEC all ones) -> WMMA-legal.
  float wl = wave_sum_f32_wmma(loss);
  float wc = wave_sum_f32_wmma(cnt);

  __shared__ float sl[8], sc[8];
  const int wave = threadIdx.x >> 5;
  const int lane = threadIdx.x & 31;
  if (lane == 0) { sl[wave] = wl; sc[wave] = wc; }
  __syncthreads();
  if (threadIdx.x == 0) {
    float L = 0.0f, C = 0.0f;
#pragma unroll
    for (int w = 0; w < 8; ++w) { L += sl[w]; C += sc[w]; }
    ws[2 * blockIdx.x + 0] = L;
    ws[2 * blockIdx.x + 1] = C;
  }
}

__global__ void __launch_bounds__(256) poly_loss_final(
    const float* __restrict__ ws, float* __restrict__ out, int nblocks)
{
  float L = 0.0f, C = 0.0f;
  for (int i = threadIdx.x; i < nblocks; i += 256) {
    L += ws[2 * i + 0];
    C += ws[2 * i + 1];
  }
  float wl = wave_sum_f32_wmma(L);
  float wc = wave_sum_f32_wmma(C);

  __shared__ float sl[8], sc[8];
  const int wave = threadIdx.x >> 5;
  const int lane = threadIdx.x & 31;
  if (lane == 0) { sl[wave] = wl; sc[wave] = wc; }
  __syncthreads();
  if (threadIdx.x == 0) {
    float Lt = 0.0f, Ct = 0.0f;
#pragma unroll
    for (int w = 0; w < 8; ++w) { Lt += sl[w]; Ct += sc[w]; }
    out[0] = Lt / fmaxf(1.0f, Ct);   // POLY_WEIGHT == 1.0
  }
}

extern "C" void kernel_launch(void* const* d_in, const int* in_sizes, int n_in,
                              void* d_out, int out_size, void* d_ws, size_t ws_size,
                              hipStream_t stream) {
  (void)in_sizes; (void)n_in; (void)out_size;
  const float* input  = (const float*)d_in[0];   // (B, 2K) f32
  const float* target = (const float*)d_in[1];   // (B, A, 4) f32
  float* out = (float*)d_out;
  float* ws  = (float*)d_ws;

  int nblocks = 2048;                            // 16 rows per thread
  size_t need = (size_t)nblocks * 2 * sizeof(float);
  if (ws_size < need) {
    nblocks = (int)(ws_size / (2 * sizeof(float)));
    if (nblocks < 1) nblocks = 1;
  }

  poly_loss_partial<<<nblocks, 256, 0, stream>>>(input, target, ws);
  poly_loss_final<<<1, 256, 0, stream>>>(ws, out, nblocks);
}